// MultiGraphSAGE_66932770341055
// MI455X (gfx1250) — compile-verified
//
#include <hip/hip_runtime.h>

#define G_ 4
#define N_ 20000
#define E_ 320000
#define C_ 128
#define H_ 128

typedef __attribute__((ext_vector_type(2))) float v2f;
typedef __attribute__((ext_vector_type(8))) float v8f;

// ---------------------------------------------------------------- zero
__global__ void zero_kernel(float* __restrict__ p, long n) {
  long i = (long)blockIdx.x * blockDim.x + threadIdx.x;
  long stride = (long)gridDim.x * blockDim.x;
  for (; i < n; i += stride) p[i] = 0.0f;
}

// ---------------------------------------------------------------- degree
__global__ void deg_kernel(const int* __restrict__ ei, float* __restrict__ deg) {
  long i = (long)blockIdx.x * blockDim.x + threadIdx.x;
  if (i >= (long)G_ * E_) return;
  int g = (int)(i / E_), e = (int)(i % E_);
  int dst = ei[(long)g * 2 * E_ + E_ + e];
  atomicAdd(&deg[(long)g * N_ + dst], 1.0f);
}

// ---------------------------------------------------------------- scatter-add of 128-wide rows
// one wave32 per edge, lane l handles features [4l, 4l+4)
__global__ __launch_bounds__(256) void scatter_kernel(const float* __restrict__ feat,
                                                      const int* __restrict__ ei,
                                                      float* __restrict__ agg) {
  long wave = (long)blockIdx.x * (blockDim.x >> 5) + (threadIdx.x >> 5);
  int lane = threadIdx.x & 31;
  if (wave >= (long)G_ * E_) return;
  int g = (int)(wave / E_), e = (int)(wave % E_);
  const int* eg = ei + (long)g * 2 * E_;
  int s = eg[e], d = eg[E_ + e];
  const float4 v = *(const float4*)(feat + ((long)g * N_ + s) * H_ + lane * 4);
  float* o = agg + ((long)g * N_ + d) * H_ + lane * 4;
  atomicAdd(o + 0, v.x);
  atomicAdd(o + 1, v.y);
  atomicAdd(o + 2, v.z);
  atomicAdd(o + 3, v.w);
}

// ---------------------------------------------------------------- fused SAGE layer GEMM:
// out[n,h] = relu( (agg[n]/max(deg,1)) . Wl[h,:] + x[n] . Wr[h,:] + bl[h] )
// Block: 256 threads = 8 waves; one 16-node tile per block; wave w owns h-slice [16w,16w+16).
__global__ __launch_bounds__(256) void sage_gemm_kernel(const float* __restrict__ agg,
                                                        const float* __restrict__ deg,
                                                        const float* __restrict__ xin,
                                                        const float* __restrict__ Wl,
                                                        const float* __restrict__ bl,
                                                        const float* __restrict__ Wr,
                                                        float* __restrict__ out) {
  __shared__ float As[16 * 132];  // mean-aggregated neighbor features (pitch 132: no bank conflicts)
  __shared__ float Xs[16 * 132];  // self features

  int tile = blockIdx.x;               // 0 .. G*N/16-1
  int g = tile / (N_ / 16);
  int row0 = (tile % (N_ / 16)) * 16;  // first node of tile
  int tid = threadIdx.x;

  const float* aggbase = agg + ((long)g * N_ + row0) * H_;
  const float* xbase = xin + ((long)g * N_ + row0) * H_;
  const float* degbase = deg + (long)g * N_ + row0;

  // cooperative stage: 512 float4s per buffer, 2 per thread; fold mean-divide into the load
  for (int i = tid; i < 512; i += 256) {
    int m = i >> 5;      // node row 0..15
    int c4 = i & 31;     // float4 column 0..31
    float inv = 1.0f / fmaxf(degbase[m], 1.0f);
    float4 a = *(const float4*)(aggbase + m * H_ + c4 * 4);
    float* dstA = &As[m * 132 + c4 * 4];
    dstA[0] = a.x * inv; dstA[1] = a.y * inv; dstA[2] = a.z * inv; dstA[3] = a.w * inv;
    float4 xv = *(const float4*)(xbase + m * H_ + c4 * 4);
    float* dstX = &Xs[m * 132 + c4 * 4];
    dstX[0] = xv.x; dstX[1] = xv.y; dstX[2] = xv.z; dstX[3] = xv.w;
  }
  __syncthreads();

  int wv = tid >> 5;          // wave 0..7
  int lane = tid & 31;
  int l16 = lane & 15;
  int kh = (lane >> 4) * 2;   // 0 for lanes 0-15, 2 for lanes 16-31
  int hbase = wv * 16;

  const float* wlrow = Wl + (long)(hbase + l16) * C_;  // B frag: B[k][n] = Wl[hbase+n][k]
  const float* wrrow = Wr + (long)(hbase + l16) * C_;
  const float* arow = &As[l16 * 132];
  const float* xrow = &Xs[l16 * 132];

  v8f acc = {0.f, 0.f, 0.f, 0.f, 0.f, 0.f, 0.f, 0.f};
  for (int kt = 0; kt < 32; ++kt) {
    int c = 4 * kt + kh;
    v2f a1 = {arow[c], arow[c + 1]};
    v2f b1 = {wlrow[c], wlrow[c + 1]};
    acc = __builtin_amdgcn_wmma_f32_16x16x4_f32(false, a1, false, b1, (short)0, acc, false, false);
    v2f a2 = {xrow[c], xrow[c + 1]};
    v2f b2 = {wrrow[c], wrrow[c + 1]};
    acc = __builtin_amdgcn_wmma_f32_16x16x4_f32(false, a2, false, b2, (short)0, acc, false, false);
  }

  // epilogue: bias + relu + store. D layout: lane<16: VGPR r = D[M=r][N=lane]; lane>=16: D[M=r+8][N=lane-16]
  float bias = bl[hbase + l16];
  int mbase = (lane >> 4) * 8;
  float* orow = out + ((long)g * N_ + row0) * H_ + hbase + l16;
#pragma unroll
  for (int r = 0; r < 8; ++r) {
    float v = fmaxf(acc[r] + bias, 0.0f);
    orow[(long)(mbase + r) * H_] = v;
  }
}

// ---------------------------------------------------------------- global max pool over all graphs/nodes
// values are post-ReLU (>=0) so integer-bit atomicMax == float max with 0-init.
__global__ __launch_bounds__(128) void pool_kernel(const float* __restrict__ h2,
                                                   float* __restrict__ pool) {
  int h = threadIdx.x;
  long row0 = (long)blockIdx.x * 128;
  long total = (long)G_ * N_;
  long rend = row0 + 128; if (rend > total) rend = total;
  float m = 0.0f;
  for (long r = row0; r < rend; ++r) m = fmaxf(m, h2[r * H_ + h]);
  atomicMax((int*)pool + h, __float_as_int(m));
}

// ---------------------------------------------------------------- tiny: Gvec MLP + fold into cvec
// gv = relu(Wg @ pool + bg); cvec[h] = bn[h] + Wn[h,128:256] . gv
__global__ __launch_bounds__(128) void gvec_kernel(const float* __restrict__ pool,
                                                   const float* __restrict__ Wg,
                                                   const float* __restrict__ bg,
                                                   const float* __restrict__ Wn,
                                                   const float* __restrict__ bn,
                                                   float* __restrict__ cvec) {
  __shared__ float pl[128];
  __shared__ float gv[128];
  int h = threadIdx.x;
  pl[h] = pool[h];
  __syncthreads();
  float s = bg[h];
  for (int c = 0; c < 128; ++c) s += Wg[h * 128 + c] * pl[c];
  gv[h] = fmaxf(s, 0.0f);
  __syncthreads();
  float t = bn[h];
  for (int c = 0; c < 128; ++c) t += Wn[h * 256 + 128 + c] * gv[c];
  cvec[h] = t;
}

// ---------------------------------------------------------------- final node MLP + output dot
// t = relu(Wn[:,0:128] @ h2row + cvec); logit = Wo . t + bo
__global__ __launch_bounds__(128) void final_kernel(const float* __restrict__ h2,
                                                    const float* __restrict__ Wn,
                                                    const float* __restrict__ cvec,
                                                    const float* __restrict__ Wo,
                                                    const float* __restrict__ bo,
                                                    float* __restrict__ out) {
  __shared__ float Wns[128 * 129];  // first half of Wn, pitch 129 (conflict-free row reads)
  __shared__ float xr[128];
  __shared__ float red[128];
  __shared__ float cv[128];
  __shared__ float wo[128];
  int tid = threadIdx.x;
  for (int i = tid; i < 128 * 128; i += 128) {
    int r = i >> 7, c = i & 127;
    Wns[r * 129 + c] = Wn[r * 256 + c];
  }
  cv[tid] = cvec[tid];
  wo[tid] = Wo[tid];
  __syncthreads();
  float bov = bo[0];

  for (long node = blockIdx.x; node < (long)G_ * N_; node += gridDim.x) {
    xr[tid] = h2[node * H_ + tid];
    __syncthreads();
    float s = cv[tid];
    const float* wrow = &Wns[tid * 129];
#pragma unroll 8
    for (int c = 0; c < 128; ++c) s += wrow[c] * xr[c];
    red[tid] = fmaxf(s, 0.0f) * wo[tid];
    __syncthreads();
    for (int off = 64; off > 0; off >>= 1) {
      if (tid < off) red[tid] += red[tid + off];
      __syncthreads();
    }
    if (tid == 0) out[node] = red[0] + bov;
    __syncthreads();
  }
}

// ---------------------------------------------------------------- launch
extern "C" void kernel_launch(void* const* d_in, const int* in_sizes, int n_in,
                              void* d_out, int out_size, void* d_ws, size_t ws_size,
                              hipStream_t stream) {
  const float* x   = (const float*)d_in[0];
  const int*   ei  = (const int*)d_in[1];
  const float* Wl1 = (const float*)d_in[2];
  const float* bl1 = (const float*)d_in[3];
  const float* Wr1 = (const float*)d_in[4];
  const float* Wl2 = (const float*)d_in[5];
  const float* bl2 = (const float*)d_in[6];
  const float* Wr2 = (const float*)d_in[7];
  const float* Wg  = (const float*)d_in[8];
  const float* bg  = (const float*)d_in[9];
  const float* Wn  = (const float*)d_in[10];
  const float* bn  = (const float*)d_in[11];
  const float* Wo  = (const float*)d_in[12];
  const float* bo  = (const float*)d_in[13];
  float* out = (float*)d_out;

  float* ws   = (float*)d_ws;
  float* agg  = ws;                                   // G*N*H  (reused in-place as layer-2 output h2)
  float* deg  = agg + (long)G_ * N_ * H_;             // G*N
  float* pool = deg + (long)G_ * N_;                  // H
  float* cvec = pool + H_;                            // H
  float* h1   = cvec + H_;                            // G*N*H

  const long aggN = (long)G_ * N_ * H_;               // 10,240,000
  const long zn1 = aggN + (long)G_ * N_ + H_;         // agg + deg + pool contiguous

  zero_kernel<<<2048, 256, 0, stream>>>(agg, zn1);
  deg_kernel<<<(G_ * E_) / 256, 256, 0, stream>>>(ei, deg);
  scatter_kernel<<<(G_ * E_) / 8, 256, 0, stream>>>(x, ei, agg);
  sage_gemm_kernel<<<(G_ * N_) / 16, 256, 0, stream>>>(agg, deg, x, Wl1, bl1, Wr1, h1);

  zero_kernel<<<2048, 256, 0, stream>>>(agg, aggN);
  scatter_kernel<<<(G_ * E_) / 8, 256, 0, stream>>>(h1, ei, agg);
  sage_gemm_kernel<<<(G_ * N_) / 16, 256, 0, stream>>>(agg, deg, h1, Wl2, bl2, Wr2, agg);  // in-place: h2 := agg

  pool_kernel<<<(G_ * N_) / 128, 128, 0, stream>>>(agg, pool);
  gvec_kernel<<<1, 128, 0, stream>>>(pool, Wg, bg, Wn, bn, cvec);
  final_kernel<<<512, 128, 0, stream>>>(agg, Wn, cvec, Wo, bo, out);
}